// InflateHexToVertex_77618648973579
// MI455X (gfx1250) — compile-verified
//
#include <hip/hip_runtime.h>

// ---- problem constants (fixed by the reference) -----------------------------
#define B_   32      // batch
#define T_   512     // num hex tiles
#define D_   128     // hex feature dim
#define V_   128     // vertex (output) dim
// K for the big conceptual GEMM is 3*D_ = 384; we split it into 3 slabs.

typedef __attribute__((ext_vector_type(2))) float v2f;
typedef __attribute__((ext_vector_type(4))) float v4f;
typedef __attribute__((ext_vector_type(8))) float v8f;

// ============================================================================
// Stage 1: P[k][b,t,v] = sum_d hex[b,t,d] * W[k*D_+d, v]
// Pure f32 WMMA (V_WMMA_F32_16X16X4_F32), exact w.r.t. reference numerics.
//   A: (B*T x D_) row-major == hex_feats flattened (contiguous, perfect).
//   B: W slab k: (D_ x V_) row-major.
//   Grid: (B*T)/16 blocks of 256 threads (8 waves). Block owns one 16-row
//   M-tile; wave w owns output columns [16w,16w+16) of ALL THREE slabs,
//   reusing each A fragment for 3 WMMAs (3 accumulators).
// ============================================================================
#define A_STRIDE 132   // 132 % 64 == 4 -> lanes m=0..15 hit distinct LDS banks

__global__ __launch_bounds__(256)
void hexvtx_stage1_wmma(const float* __restrict__ hex,
                        const float* __restrict__ W,
                        float* __restrict__ P)
{
    const int mtile = blockIdx.x;            // 0 .. B_*T_/16 - 1
    const int tid   = threadIdx.x;
    const int wave  = tid >> 5;              // 0..7
    const int lane  = tid & 31;

    __shared__ float Alds[16 * A_STRIDE];    // 16 x 128 A-tile, padded rows

    // Cooperative A-tile load: 16 rows x 128 f32 = 512 float4s / 256 threads.
    {
        const float* Abase = hex + (size_t)mtile * 16 * D_;
        for (int i = tid; i < 16 * (D_ / 4); i += 256) {
            const int r  = i >> 5;           // i / 32 rows of 32 float4s
            const int c4 = i & 31;           // float4 column
            v4f v = *(const v4f*)(Abase + (size_t)r * D_ + c4 * 4);
            *(v4f*)(&Alds[r * A_STRIDE + c4 * 4]) = v;   // 16B aligned (132*4=528=33*16)
        }
    }
    __syncthreads();

    // WMMA f32 16x16x4 fragment addressing (ISA 7.12.2, 32-bit A 16x4):
    //   lane L: m = L&15 ; holds K = koff, koff+1 where koff = (L>=16)*2
    const int m    = lane & 15;
    const int koff = (lane >> 4) << 1;       // 0 or 2
    const int n    = lane & 15;              // B/C/D column within tile

    const int col = wave * 16 + n;           // output column 0..127
    const float* W0 = W + (size_t)(0 * D_) * V_ + col;
    const float* W1 = W + (size_t)(1 * D_) * V_ + col;
    const float* W2 = W + (size_t)(2 * D_) * V_ + col;

    v8f acc0 = {}, acc1 = {}, acc2 = {};

    #pragma unroll 4
    for (int kk = 0; kk < D_; kk += 4) {
        const int r = kk + koff;
        v2f a;
        a.x = Alds[m * A_STRIDE + r];
        a.y = Alds[m * A_STRIDE + r + 1];

        v2f b0, b1, b2;
        b0.x = W0[(size_t)r * V_];  b0.y = W0[(size_t)(r + 1) * V_];
        b1.x = W1[(size_t)r * V_];  b1.y = W1[(size_t)(r + 1) * V_];
        b2.x = W2[(size_t)r * V_];  b2.y = W2[(size_t)(r + 1) * V_];

        // 8-arg pattern: (neg_a, A, neg_b, B, c_mod, C, reuse_a, reuse_b)
        acc0 = __builtin_amdgcn_wmma_f32_16x16x4_f32(false, a, false, b0,
                                                     (short)0, acc0, false, false);
        acc1 = __builtin_amdgcn_wmma_f32_16x16x4_f32(false, a, false, b1,
                                                     (short)0, acc1, false, false);
        acc2 = __builtin_amdgcn_wmma_f32_16x16x4_f32(false, a, false, b2,
                                                     (short)0, acc2, false, false);
    }

    // D layout (32-bit C/D 16x16): VGPR j -> M = j (lanes 0-15) / j+8 (16-31),
    // N = lane&15.
    const size_t MTV = (size_t)B_ * T_ * V_;
    const int mhi = (lane >> 4) << 3;        // 0 or 8
    #pragma unroll
    for (int j = 0; j < 8; ++j) {
        const int    mrow = j + mhi;
        const size_t off  = ((size_t)mtile * 16 + mrow) * V_ + wave * 16 + (lane & 15);
        P[0 * MTV + off] = acc0[j];
        P[1 * MTV + off] = acc1[j];
        P[2 * MTV + off] = acc2[j];
    }
}

// ============================================================================
// Stage 2: out[b,n,v] = bias[v] + sum_k (idx[n,k]>=0 ? P[k][b,idx[n,k],v] : 0)
// One block per vertex; 8 groups of 32 lanes sweep the 32 batches; lane owns
// a float4 of V (32 lanes * 4 = 128 = V_). P reads hit L2 (25 MB resident);
// output stream (327 MB) written non-temporally to avoid evicting P.
// ============================================================================
__global__ __launch_bounds__(256)
void hexvtx_stage2_gather(const float* __restrict__ P,
                          const int*   __restrict__ v2h,
                          const float* __restrict__ bias,
                          float* __restrict__ out,
                          int nverts)
{
    const int nv   = blockIdx.x;
    const int grp  = threadIdx.x >> 5;       // 0..7
    const int lane = threadIdx.x & 31;
    const int v    = lane << 2;              // float4 column

    const int t0 = v2h[(size_t)nv * 3 + 0];
    const int t1 = v2h[(size_t)nv * 3 + 1];
    const int t2 = v2h[(size_t)nv * 3 + 2];

    const v4f bb = *(const v4f*)(bias + v);
    const size_t MTV = (size_t)B_ * T_ * V_;

    #pragma unroll
    for (int b = grp; b < B_; b += 8) {
        v4f acc = bb;
        if (t0 >= 0) acc += *(const v4f*)(P + 0 * MTV + ((size_t)b * T_ + t0) * V_ + v);
        if (t1 >= 0) acc += *(const v4f*)(P + 1 * MTV + ((size_t)b * T_ + t1) * V_ + v);
        if (t2 >= 0) acc += *(const v4f*)(P + 2 * MTV + ((size_t)b * T_ + t2) * V_ + v);
        v4f* dst = (v4f*)(out + ((size_t)b * nverts + nv) * V_ + v);
        __builtin_nontemporal_store(acc, dst);
    }
}

// ============================================================================
// Fallback (only if workspace is too small for P): direct gathered dot.
// One block of 128 threads per (b, n); thread v computes one output element.
// ============================================================================
__global__ __launch_bounds__(128)
void hexvtx_fallback(const float* __restrict__ hex,
                     const int*   __restrict__ v2h,
                     const float* __restrict__ W,
                     const float* __restrict__ bias,
                     float* __restrict__ out,
                     int nverts)
{
    const size_t bn = blockIdx.x;            // 0 .. B_*nverts - 1
    const int b  = (int)(bn / nverts);
    const int nv = (int)(bn % nverts);
    const int v  = threadIdx.x;

    float acc = bias[v];
    #pragma unroll
    for (int k = 0; k < 3; ++k) {
        const int t = v2h[(size_t)nv * 3 + k];
        if (t >= 0) {
            const float* hrow = hex + ((size_t)b * T_ + t) * D_;
            const float* wcol = W + (size_t)(k * D_) * V_ + v;
            for (int d = 0; d < D_; ++d)
                acc = fmaf(hrow[d], wcol[(size_t)d * V_], acc);
        }
    }
    out[bn * V_ + v] = acc;
}

// ============================================================================
extern "C" void kernel_launch(void* const* d_in, const int* in_sizes, int n_in,
                              void* d_out, int out_size, void* d_ws, size_t ws_size,
                              hipStream_t stream)
{
    (void)n_in; (void)out_size;
    const float* hex  = (const float*)d_in[0];   // (B,T,D) f32
    const int*   v2h  = (const int*)  d_in[1];   // (N,3)   int
    const float* W    = (const float*)d_in[2];   // (3D, V) f32
    const float* bias = (const float*)d_in[3];   // (V,)    f32
    float*       out  = (float*)d_out;           // (B,N,V) f32

    const int nverts = in_sizes[1] / 3;
    const size_t need = 3ull * B_ * T_ * V_ * sizeof(float);   // 25.2 MB

    if (ws_size >= need) {
        float* P = (float*)d_ws;
        hipLaunchKernelGGL(hexvtx_stage1_wmma,
                           dim3((B_ * T_) / 16), dim3(256), 0, stream,
                           hex, W, P);
        hipLaunchKernelGGL(hexvtx_stage2_gather,
                           dim3(nverts), dim3(256), 0, stream,
                           P, v2h, bias, out, nverts);
    } else {
        hipLaunchKernelGGL(hexvtx_fallback,
                           dim3((unsigned)((size_t)B_ * nverts)), dim3(V_), 0, stream,
                           hex, v2h, W, bias, out, nverts);
    }
}